// Qwen3VisionAttention_80848464380507
// MI455X (gfx1250) — compile-verified
//
#include <hip/hip_runtime.h>
#include <math.h>

#define DIM 1280
#define HEADS 16
#define HEAD_DIM 80
#define NCHUNKS 8
#define CHUNK 1024
#define TOTAL (NCHUNKS * CHUNK)
#define QKV_N (3 * DIM)

typedef __attribute__((ext_vector_type(2))) float v2f;
typedef __attribute__((ext_vector_type(8))) float v8f;
typedef __attribute__((ext_vector_type(4))) int v4i;

static __device__ __forceinline__ v8f wmma_f32(v2f a, v2f b, v8f c) {
  // D = A(16x4) * B(4x16) + C(16x16), full fp32
  return __builtin_amdgcn_wmma_f32_16x16x4_f32(false, a, false, b, (short)0, c,
                                               false, false);
}

// -------- CDNA5 async global->LDS copy (ASYNCcnt-tracked), with fallback ----
#ifndef USE_ASYNC_LDS
#if defined(__has_builtin)
#if __has_builtin(__builtin_amdgcn_global_load_async_to_lds_b128)
#define USE_ASYNC_LDS 1
#endif
#endif
#endif
#ifndef USE_ASYNC_LDS
#define USE_ASYNC_LDS 0
#endif

#if USE_ASYNC_LDS
typedef v4i __attribute__((address_space(1)))* gv4p;  // global int4*
typedef v4i __attribute__((address_space(3)))* lv4p;  // LDS int4*
static __device__ __forceinline__ void async_copy16(const float* g, float* l) {
  __builtin_amdgcn_global_load_async_to_lds_b128((gv4p)g, (lv4p)l, 0, 0);
}
static __device__ __forceinline__ void wait_async0() {
#if defined(__has_builtin) && __has_builtin(__builtin_amdgcn_s_wait_asynccnt)
  __builtin_amdgcn_s_wait_asynccnt(0);
#else
  asm volatile("s_wait_asynccnt 0x0" ::: "memory");
#endif
}
#endif

// ---------------------------------------------------------------------------
// GEMM: C[M][N] = A[M][K] * W[N][K]^T + bias[N]
// 256 threads = 8 waves. Macro tile 128(M) x 64(N). Wave w: rows [16w,16w+16),
// all 4 N-tiles. Double-buffered LDS, K slabs of 32, next-slab fetch
// overlapped with WMMA compute (async-to-LDS when available).
// ---------------------------------------------------------------------------
#define GBM 128
#define GBN 64
#define GBK 32

__global__ __launch_bounds__(256) void gemm_xwt_bias(
    const float* __restrict__ A, const float* __restrict__ W,
    const float* __restrict__ bias, float* __restrict__ C, int M, int N,
    int K) {
  __shared__ float sA[2][GBM][GBK];  // 32 KB
  __shared__ float sW[2][GBN][GBK];  // 16 KB

  const int bm = blockIdx.y * GBM;
  const int bn = blockIdx.x * GBN;
  const int tid = threadIdx.x;
  const int wave = tid >> 5;
  const int lane = tid & 31;
  const int half = lane >> 4;  // 0: K=k,k+1  1: K=k+2,k+3
  const int l16 = lane & 15;
  const int half2 = half * 2;
  const int nslab = K / GBK;

  v8f acc[4] = {};

#if USE_ASYNC_LDS
  auto issue_slab = [&](int k0, int buf) {
#pragma unroll
    for (int j = 0; j < 4; ++j) {  // A: 1024 vec4 slots
      int i = tid + j * 256, r = i >> 3, c4 = (i & 7) * 4;
      async_copy16(&A[(size_t)(bm + r) * K + k0 + c4], &sA[buf][r][c4]);
    }
#pragma unroll
    for (int j = 0; j < 2; ++j) {  // W: 512 vec4 slots
      int i = tid + j * 256, r = i >> 3, c4 = (i & 7) * 4;
      async_copy16(&W[(size_t)(bn + r) * K + k0 + c4], &sW[buf][r][c4]);
    }
  };
  issue_slab(0, 0);
  wait_async0();
  __syncthreads();
#else
  float4 ra[4], rw[2];
  auto load_slab = [&](int k0) {
#pragma unroll
    for (int j = 0; j < 4; ++j) {
      int i = tid + j * 256, r = i >> 3, c4 = (i & 7) * 4;
      ra[j] = *(const float4*)&A[(size_t)(bm + r) * K + k0 + c4];
    }
#pragma unroll
    for (int j = 0; j < 2; ++j) {
      int i = tid + j * 256, r = i >> 3, c4 = (i & 7) * 4;
      rw[j] = *(const float4*)&W[(size_t)(bn + r) * K + k0 + c4];
    }
  };
  auto store_slab = [&](int buf) {
#pragma unroll
    for (int j = 0; j < 4; ++j) {
      int i = tid + j * 256, r = i >> 3, c4 = (i & 7) * 4;
      *(float4*)&sA[buf][r][c4] = ra[j];
    }
#pragma unroll
    for (int j = 0; j < 2; ++j) {
      int i = tid + j * 256, r = i >> 3, c4 = (i & 7) * 4;
      *(float4*)&sW[buf][r][c4] = rw[j];
    }
  };
  load_slab(0);
  store_slab(0);
  __syncthreads();
#endif

  for (int it = 0; it < nslab; ++it) {
    const int cur = it & 1;
#if USE_ASYNC_LDS
    if (it + 1 < nslab) issue_slab((it + 1) * GBK, cur ^ 1);
#else
    if (it + 1 < nslab) load_slab((it + 1) * GBK);
#endif
    const int arow = wave * 16 + l16;
#pragma unroll
    for (int kk = 0; kk < GBK; kk += 4) {
      v2f af = *(const v2f*)&sA[cur][arow][kk + half2];
#pragma unroll
      for (int t = 0; t < 4; ++t) {
        v2f bf = *(const v2f*)&sW[cur][t * 16 + l16][kk + half2];
        acc[t] = wmma_f32(af, bf, acc[t]);
      }
    }
#if USE_ASYNC_LDS
    wait_async0();
#else
    if (it + 1 < nslab) store_slab(cur ^ 1);
#endif
    __syncthreads();
  }

  // C/D layout: VGPR v -> row v (lanes 0-15) or v+8 (lanes 16-31), col = l16
#pragma unroll
  for (int t = 0; t < 4; ++t) {
    int col = bn + t * 16 + l16;
    float b = bias[col];
#pragma unroll
    for (int v = 0; v < 8; ++v) {
      int row = bm + wave * 16 + (half ? v + 8 : v);
      C[(size_t)row * N + col] = acc[t][v] + b;
    }
  }
}

// ---------------------------------------------------------------------------
// RoPE in place on Q and K parts of the QKV buffer.
// ---------------------------------------------------------------------------
__global__ void rope_kernel(float* __restrict__ qkv,
                            const float* __restrict__ cosb,
                            const float* __restrict__ sinb) {
  int idx = blockIdx.x * blockDim.x + threadIdx.x;
  const int NPAIR = TOTAL * HEADS * (HEAD_DIM / 2);
  if (idx >= NPAIR) return;
  int p = idx % (HEAD_DIM / 2);
  int t = idx / (HEAD_DIM / 2);
  int h = t % HEADS;
  int tok = t / HEADS;

  float c1 = cosb[tok * HEAD_DIM + p];
  float s1 = sinb[tok * HEAD_DIM + p];
  float c2 = cosb[tok * HEAD_DIM + p + 40];
  float s2 = sinb[tok * HEAD_DIM + p + 40];

  size_t base = (size_t)tok * QKV_N + h * HEAD_DIM + p;
  {
    float x1 = qkv[base], x2 = qkv[base + 40];
    qkv[base] = x1 * c1 - x2 * s1;
    qkv[base + 40] = x2 * c2 + x1 * s2;
  }
  {
    size_t kb = base + DIM;
    float x1 = qkv[kb], x2 = qkv[kb + 40];
    qkv[kb] = x1 * c1 - x2 * s1;
    qkv[kb + 40] = x2 * c2 + x1 * s2;
  }
}

// ---------------------------------------------------------------------------
// Flash attention. Grid: (row_blocks=8, heads=16, chunks=8). 256 threads.
// 128 query rows resident in LDS (pre-scaled); stream 64-key blocks with the
// next K/V tile register-staged under the current tile's compute.
// ---------------------------------------------------------------------------
#define AQ 128
#define AKB 64
#define HDP 84
#define PKP 68

__global__ __launch_bounds__(256) void attn_kernel(
    const float* __restrict__ qkv, float* __restrict__ attn) {
  __shared__ float sQ[AQ][HDP];
  __shared__ float sK[AKB][HDP];
  __shared__ float sV[AKB][HDP];
  __shared__ float sP[8][16][PKP];

  const int rb = blockIdx.x;
  const int h = blockIdx.y;
  const int c = blockIdx.z;
  const int tid = threadIdx.x;
  const int wave = tid >> 5;
  const int lane = tid & 31;
  const int half = lane >> 4;
  const int l16 = lane & 15;
  const int half2 = half * 2;

  const int base_tok = c * CHUNK + rb * AQ;
  const float scale = 0.11180339887498948f;  // 80^-0.5

  // stage Q tile, pre-multiplied by softmax scale
  for (int i = tid; i < AQ * (HEAD_DIM / 4); i += 256) {
    int r = i / (HEAD_DIM / 4);
    int c4 = (i % (HEAD_DIM / 4)) * 4;
    float4 v = *(const float4*)&qkv[(size_t)(base_tok + r) * QKV_N +
                                    h * HEAD_DIM + c4];
    v.x *= scale; v.y *= scale; v.z *= scale; v.w *= scale;
    *(float4*)&sQ[r][c4] = v;
  }

  // K/V pipeline: register-stage next 64-key tile under current compute
  float4 rk[5], rv[5];
  auto issue_kv = [&](int kb) {
    const int ktok = c * CHUNK + kb * AKB;
#pragma unroll
    for (int j = 0; j < 5; ++j) {
      int i = tid + j * 256;
      int r = i / (HEAD_DIM / 4);
      int c4 = (i % (HEAD_DIM / 4)) * 4;
      size_t rowoff = (size_t)(ktok + r) * QKV_N + h * HEAD_DIM + c4;
      rk[j] = *(const float4*)&qkv[rowoff + DIM];
      rv[j] = *(const float4*)&qkv[rowoff + 2 * DIM];
    }
  };
  auto store_kv = [&]() {
#pragma unroll
    for (int j = 0; j < 5; ++j) {
      int i = tid + j * 256;
      int r = i / (HEAD_DIM / 4);
      int c4 = (i % (HEAD_DIM / 4)) * 4;
      *(float4*)&sK[r][c4] = rk[j];
      *(float4*)&sV[r][c4] = rv[j];
    }
  };

  v8f o[5] = {};
  float m_i[8], l_i[8];
#pragma unroll
  for (int v = 0; v < 8; ++v) {
    m_i[v] = -INFINITY;
    l_i[v] = 0.0f;
  }

  issue_kv(0);
  for (int kb = 0; kb < CHUNK / AKB; ++kb) {
    store_kv();                                  // commit tile kb (waits loads)
    if (kb + 1 < CHUNK / AKB) issue_kv(kb + 1);  // fetch kb+1 under compute
    __syncthreads();

    // S = (Q*scale) . K^T : 16 rows x 64 keys per wave
    v8f s[4] = {};
    const int arow = wave * 16 + l16;
#pragma unroll
    for (int kk = 0; kk < HEAD_DIM; kk += 4) {
      v2f af = *(const v2f*)&sQ[arow][kk + half2];
#pragma unroll
      for (int t = 0; t < 4; ++t) {
        v2f bf = *(const v2f*)&sK[t * 16 + l16][kk + half2];
        s[t] = wmma_f32(af, bf, s[t]);
      }
    }

    // online softmax (rows live in half-waves; 16-lane shfl reductions)
#pragma unroll
    for (int v = 0; v < 8; ++v) {
      float tm = fmaxf(fmaxf(s[0][v], s[1][v]), fmaxf(s[2][v], s[3][v]));
#pragma unroll
      for (int mask = 1; mask <= 8; mask <<= 1)
        tm = fmaxf(tm, __shfl_xor(tm, mask, 32));
      float mnew = fmaxf(m_i[v], tm);
      float corr = __expf(m_i[v] - mnew);
      int prow = half ? v + 8 : v;
      float rs = 0.0f;
#pragma unroll
      for (int t = 0; t < 4; ++t) {
        float pj = __expf(s[t][v] - mnew);
        sP[wave][prow][t * 16 + l16] = pj;
        rs += pj;
      }
#pragma unroll
      for (int mask = 1; mask <= 8; mask <<= 1)
        rs += __shfl_xor(rs, mask, 32);
      l_i[v] = l_i[v] * corr + rs;
      m_i[v] = mnew;
#pragma unroll
      for (int d = 0; d < 5; ++d) o[d][v] *= corr;
    }

    // O += P(16x64) . V(64x80); P reloaded from LDS in A-fragment layout
#pragma unroll
    for (int kk = 0; kk < AKB; kk += 4) {
      v2f af = *(const v2f*)&sP[wave][l16][kk + half2];
#pragma unroll
      for (int t = 0; t < 5; ++t) {
        v2f bf;
        bf.x = sV[kk + half2 + 0][t * 16 + l16];
        bf.y = sV[kk + half2 + 1][t * 16 + l16];
        o[t] = wmma_f32(af, bf, o[t]);
      }
    }
    __syncthreads();  // all waves done with sK/sV before next commit
  }

  // normalize and write [token][head*80 + d]
#pragma unroll
  for (int v = 0; v < 8; ++v) {
    float inv = 1.0f / l_i[v];
    int row = base_tok + wave * 16 + (half ? v + 8 : v);
#pragma unroll
    for (int t = 0; t < 5; ++t)
      attn[(size_t)row * DIM + h * HEAD_DIM + t * 16 + l16] = o[t][v] * inv;
  }
}

// ---------------------------------------------------------------------------
extern "C" void kernel_launch(void* const* d_in, const int* in_sizes, int n_in,
                              void* d_out, int out_size, void* d_ws,
                              size_t ws_size, hipStream_t stream) {
  const float* hidden = (const float*)d_in[0];
  const float* cosb = (const float*)d_in[1];
  const float* sinb = (const float*)d_in[2];
  const float* qkvw = (const float*)d_in[3];
  const float* qkvb = (const float*)d_in[4];
  const float* projw = (const float*)d_in[5];
  const float* projb = (const float*)d_in[6];
  float* out = (float*)d_out;

  float* qkv = (float*)d_ws;                  // TOTAL x 3840 f32 (126 MB)
  float* attn = qkv + (size_t)TOTAL * QKV_N;  // TOTAL x 1280 f32 (42 MB)

  dim3 blk(256);

  gemm_xwt_bias<<<dim3(QKV_N / GBN, TOTAL / GBM), blk, 0, stream>>>(
      hidden, qkvw, qkvb, qkv, TOTAL, QKV_N, DIM);

  {
    int npair = TOTAL * HEADS * (HEAD_DIM / 2);
    rope_kernel<<<(npair + 255) / 256, blk, 0, stream>>>(qkv, cosb, sinb);
  }

  attn_kernel<<<dim3(CHUNK / AQ, HEADS, NCHUNKS), blk, 0, stream>>>(qkv, attn);

  gemm_xwt_bias<<<dim3(DIM / GBN, TOTAL / GBM), blk, 0, stream>>>(
      attn, projw, projb, out, TOTAL, DIM, DIM);
}